// SVDO_68324339745368
// MI455X (gfx1250) — compile-verified
//
#include <hip/hip_runtime.h>
#include <hip/hip_bf16.h>
#include <stdint.h>

#define THREADS 256
#define MPB 256                 // matrices per block
#define TILE_F (MPB * 9)        // 2304 floats = 9216 bytes staged per block

typedef unsigned int       u32;
typedef unsigned long long u64;
typedef u32 v4u __attribute__((ext_vector_type(4)));
typedef int v4i __attribute__((ext_vector_type(4)));
typedef int v8i __attribute__((ext_vector_type(8)));

#if defined(__gfx1250__) && __has_builtin(__builtin_amdgcn_tensor_load_to_lds) && \
    __has_builtin(__builtin_amdgcn_tensor_store_from_lds)
#define USE_TDM 1
#else
#define USE_TDM 0
#endif

#if USE_TDM
// Build the D# for a 1-row tile of `tile0` 4-byte elements at global byte addr `gaddr`,
// LDS byte offset `ldsoff`, with `tdim0` elements valid from tile start (OOB guard).
__device__ __forceinline__ void tdm_desc(v4u& g0, v8i& g1, u64 gaddr, u32 ldsoff,
                                         u32 tdim0, u32 tile0)
{
    // group 0 (128b): count=1 | lds_addr | global_addr[56:0] | type=2
    g0[0] = 1u;
    g0[1] = ldsoff;
    g0[2] = (u32)(gaddr & 0xFFFFFFFFull);
    g0[3] = (u32)((gaddr >> 32) & 0x01FFFFFFull) | 0x80000000u;
    // group 1 (256b): data_size=4B; tensor_dim0=tdim0; tensor_dim1=1; tile_dim0=tile0; tile_dim1=1
    g1[0] = (int)(2u << 16);                               // data_size = 4 bytes
    g1[1] = (int)((tdim0 & 0xFFFFu) << 16);                // tensor_dim0[15:0] -> bits 63:48
    g1[2] = (int)(((tdim0 >> 16) & 0xFFFFu) | (1u << 16)); // tensor_dim0[31:16]; tensor_dim1=1
    g1[3] = (int)(tile0 << 16);                            // tile_dim0 -> bits 127:112
    g1[4] = 1;                                             // tile_dim1=1, tile_dim2=0
    g1[5] = (int)tile0;                                    // tensor_dim0_stride[31:0] (single row)
    g1[6] = 0;
    g1[7] = 0;
}

__device__ __forceinline__ void tdm_wait0()
{
#if __has_builtin(__builtin_amdgcn_s_wait_tensorcnt)
    __builtin_amdgcn_s_wait_tensorcnt(0);
#else
    asm volatile("s_wait_tensorcnt 0x0" ::: "memory");
#endif
}
#endif // USE_TDM

// One Jacobi rotation on symmetric S (pair p,q; r = third index), accumulate V <- V*J.
__device__ __forceinline__ void jrot(float S[3][3], float V[3][3], int p, int q, int r)
{
    float apq = S[p][q];
    float c = 1.0f, s = 0.0f, t = 0.0f;
    if (fabsf(apq) > 1e-20f) {
        float tau = (S[q][q] - S[p][p]) * (0.5f / apq);
        t = copysignf(1.0f, tau) / (fabsf(tau) + sqrtf(1.0f + tau * tau));
        c = rsqrtf(1.0f + t * t);
        s = t * c;
    }
    float app = S[p][p], aqq = S[q][q], apr = S[p][r], aqr = S[q][r];
    S[p][p] = app - t * apq;
    S[q][q] = aqq + t * apq;
    S[p][q] = 0.0f; S[q][p] = 0.0f;
    float npr = c * apr - s * aqr;
    float nqr = s * apr + c * aqr;
    S[p][r] = npr; S[r][p] = npr;
    S[q][r] = nqr; S[r][q] = nqr;
#pragma unroll
    for (int k = 0; k < 3; ++k) {
        float vp = V[k][p], vq = V[k][q];
        V[k][p] = c * vp - s * vq;
        V[k][q] = s * vp + c * vq;
    }
}

__device__ __forceinline__ void swapcols(float V[3][3], int a, int b)
{
#pragma unroll
    for (int k = 0; k < 3; ++k) { float t = V[k][a]; V[k][a] = V[k][b]; V[k][b] = t; }
}

__global__ __launch_bounds__(THREADS)
void so3_orth_kernel(const float* __restrict__ x, float* __restrict__ out, int nmat)
{
    __shared__ float tile[TILE_F];
    const int tid = threadIdx.x;
    const long long base9  = (long long)blockIdx.x * TILE_F;
    const long long total9 = (long long)nmat * 9ll;
    const long long remll  = total9 - base9;
    const int nelem = (remll > (long long)TILE_F) ? TILE_F : (int)remll;

    // ---------------- stage input tile: global -> LDS via Tensor Data Mover ----------------
#if USE_TDM
    if (tid < 32) {   // wave 0 issues a single DMA for the whole block's tile
        v4u g0; v8i g1;
        tdm_desc(g0, g1,
                 (u64)(uintptr_t)(x + base9),
                 (u32)(uintptr_t)(&tile[0]),    // low 32 bits of flat addr = LDS byte offset
                 (u32)remll, (u32)nelem);
        v4i g2; g2[0]=0; g2[1]=0; g2[2]=0; g2[3]=0;
        v4i g3; g3[0]=0; g3[1]=0; g3[2]=0; g3[3]=0;
#if defined(__clang_major__) && (__clang_major__ >= 23)
        v8i g4; g4[0]=0; g4[1]=0; g4[2]=0; g4[3]=0; g4[4]=0; g4[5]=0; g4[6]=0; g4[7]=0;
        __builtin_amdgcn_tensor_load_to_lds(g0, g1, g2, g3, g4, 0);
#else
        __builtin_amdgcn_tensor_load_to_lds(g0, g1, g2, g3, 0);
#endif
        tdm_wait0();
    }
#else
    // Fallback: cooperative coalesced 128-bit loads into LDS
    {
        const float4* __restrict__ g4p = (const float4*)(x + base9);
        const int n4 = nelem >> 2;
        for (int i = tid; i < n4; i += THREADS) ((float4*)tile)[i] = g4p[i];
        for (int i = (n4 << 2) + tid; i < nelem; i += THREADS) tile[i] = x[base9 + i];
    }
#endif
    __syncthreads();

    // ---------------- per-thread 3x3 SO(3) projection ----------------
    const int gmat = blockIdx.x * MPB + tid;
    if (gmat < nmat) {
        float m[3][3];
#pragma unroll
        for (int i = 0; i < 3; ++i)
#pragma unroll
            for (int j = 0; j < 3; ++j)
                m[i][j] = tile[tid * 9 + i * 3 + j];   // stride-9 dwords: bank-conflict-free

        // S = M^T M (symmetric)
        float S[3][3];
#pragma unroll
        for (int i = 0; i < 3; ++i)
#pragma unroll
            for (int j = 0; j < 3; ++j)
                S[i][j] = m[0][i]*m[0][j] + m[1][i]*m[1][j] + m[2][i]*m[2][j];

        float V[3][3] = {{1,0,0},{0,1,0},{0,0,1}};     // det(V) stays +1 (rotations only)
#pragma unroll
        for (int sweep = 0; sweep < 3; ++sweep) {
            jrot(S, V, 0, 1, 2);
            jrot(S, V, 0, 2, 1);
            jrot(S, V, 1, 2, 0);
        }

        // sort eigenvalues descending, keep det(V) = +1 via parity fix on last column
        float d0 = S[0][0], d1 = S[1][1], d2 = S[2][2];
        bool flip = false;
        if (d0 < d1) { float t=d0; d0=d1; d1=t; swapcols(V,0,1); flip=!flip; }
        if (d1 < d2) { float t=d1; d1=d2; d2=t; swapcols(V,1,2); flip=!flip; }
        if (d0 < d1) { float t=d0; d0=d1; d1=t; swapcols(V,0,1); flip=!flip; }
        if (flip) { V[0][2]=-V[0][2]; V[1][2]=-V[1][2]; V[2][2]=-V[2][2]; }

        // U from Gram-Schmidt of M*v0, M*v1; u2 = u0 x u1  =>  det(U)=+1
        float b0[3], b1[3];
#pragma unroll
        for (int i = 0; i < 3; ++i) {
            b0[i] = m[i][0]*V[0][0] + m[i][1]*V[1][0] + m[i][2]*V[2][0];
            b1[i] = m[i][0]*V[0][1] + m[i][1]*V[1][1] + m[i][2]*V[2][1];
        }
        float inv0 = rsqrtf(fmaxf(b0[0]*b0[0]+b0[1]*b0[1]+b0[2]*b0[2], 1e-24f));
        float u0[3] = { b0[0]*inv0, b0[1]*inv0, b0[2]*inv0 };
        float proj = u0[0]*b1[0] + u0[1]*b1[1] + u0[2]*b1[2];
        float w[3] = { b1[0]-proj*u0[0], b1[1]-proj*u0[1], b1[2]-proj*u0[2] };
        float inv1 = rsqrtf(fmaxf(w[0]*w[0]+w[1]*w[1]+w[2]*w[2], 1e-24f));
        float u1[3] = { w[0]*inv1, w[1]*inv1, w[2]*inv1 };
        float u2[3] = { u0[1]*u1[2]-u0[2]*u1[1],
                        u0[2]*u1[0]-u0[0]*u1[2],
                        u0[0]*u1[1]-u0[1]*u1[0] };

        // R = U V^T  (== U diag(1,1,det(UV^T)) V^T of the reference)
#pragma unroll
        for (int i = 0; i < 3; ++i)
#pragma unroll
            for (int j = 0; j < 3; ++j)
                tile[tid*9 + i*3 + j] = u0[i]*V[j][0] + u1[i]*V[j][1] + u2[i]*V[j][2];
    }
    __syncthreads();   // drains DScnt: the result tile is fully visible in LDS

    // ---------------- drain output tile: LDS -> global via Tensor Data Mover ----------------
#if USE_TDM
    if (tid < 32) {   // wave 0 issues a single DMA store for the whole block's tile
        v4u g0; v8i g1;
        tdm_desc(g0, g1,
                 (u64)(uintptr_t)(out + base9),
                 (u32)(uintptr_t)(&tile[0]),
                 (u32)remll, (u32)nelem);     // writes beyond tensor_dim0 are dropped
        v4i g2; g2[0]=0; g2[1]=0; g2[2]=0; g2[3]=0;
        v4i g3; g3[0]=0; g3[1]=0; g3[2]=0; g3[3]=0;
#if defined(__clang_major__) && (__clang_major__ >= 23)
        v8i g4; g4[0]=0; g4[1]=0; g4[2]=0; g4[3]=0; g4[4]=0; g4[5]=0; g4[6]=0; g4[7]=0;
        __builtin_amdgcn_tensor_store_from_lds(g0, g1, g2, g3, g4, 0);
#else
        __builtin_amdgcn_tensor_store_from_lds(g0, g1, g2, g3, 0);
#endif
        tdm_wait0();  // LDS/workgroup stays alive until issuing wave drains TENSORcnt
    }
#else
    // Fallback: cooperative coalesced 128-bit stores from LDS
    {
        float4* __restrict__ o4 = (float4*)(out + base9);
        const int n4 = nelem >> 2;
        for (int i = tid; i < n4; i += THREADS) o4[i] = ((const float4*)tile)[i];
        for (int i = (n4 << 2) + tid; i < nelem; i += THREADS) out[base9 + i] = tile[i];
    }
#endif
}

extern "C" void kernel_launch(void* const* d_in, const int* in_sizes, int n_in,
                              void* d_out, int out_size, void* d_ws, size_t ws_size,
                              hipStream_t stream)
{
    const float* x = (const float*)d_in[0];
    float* out = (float*)d_out;
    const int nmat = in_sizes[0] / 9;
    const int blocks = (nmat + MPB - 1) / MPB;
    so3_orth_kernel<<<blocks, THREADS, 0, stream>>>(x, out, nmat);
}